// YOLOv5Loss_55843164783171
// MI455X (gfx1250) — compile-verified
//
#include <hip/hip_runtime.h>
#include <cmath>

typedef __attribute__((ext_vector_type(2))) float v2f;
typedef __attribute__((ext_vector_type(8))) float v8f;

#define NT_MAX 300
#define ENT_MAX (15 * NT_MAX) /* 4500 */
#define CPART_MAX 256

static __device__ __forceinline__ float bce_f(float x, float t) {
    return fmaxf(x, 0.0f) - x * t + log1pf(expf(-fabsf(x)));
}
static __device__ __forceinline__ float bce0_f(float x) {
    return fmaxf(x, 0.0f) + log1pf(expf(-fabsf(x)));
}
static __device__ __forceinline__ float sigm_f(float x) {
    return 1.0f / (1.0f + expf(-x));
}

// Deterministic block reduction; result valid on thread 0.
static __device__ float block_sum(float v, float* sm) {
    for (int o = 16; o > 0; o >>= 1) v += __shfl_down(v, o, 32);
    int wid = threadIdx.x >> 5;
    int lane = threadIdx.x & 31;
    __syncthreads();
    if (lane == 0) sm[wid] = v;
    __syncthreads();
    float r = 0.0f;
    if (threadIdx.x == 0) {
        int nw = blockDim.x >> 5;
        for (int i = 0; i < nw; ++i) r += sm[i];
    }
    return r;
}

// ---------------------------------------------------------------------------
// Kernel A: per-(offset, anchor, target) assignment, CIoU, class BCE.
// ---------------------------------------------------------------------------
__global__ void assignKernel(const float* __restrict__ pred,
                             const float* __restrict__ targets,
                             int* __restrict__ eCell, float* __restrict__ eVal,
                             float* __restrict__ ePm4, float* __restrict__ aPart,
                             int NT, int W, int H,
                             float a0x, float a0y, float a1x, float a1y,
                             float a2x, float a2y)
{
    __shared__ float sm[8];
    int e = blockIdx.x * blockDim.x + threadIdx.x;
    int nEnt = 15 * NT;
    float lbox_c = 0.0f, cnt_c = 0.0f, lcls_c = 0.0f;
    if (e < nEnt) {
        int off = e / (3 * NT);
        int rem = e - off * 3 * NT;
        int an  = rem / NT;
        int tt  = rem - an * NT;
        const float* tg = targets + tt * 6;
        float ax = (an == 0) ? a0x : (an == 1) ? a1x : a2x;
        float ay = (an == 0) ? a0y : (an == 1) ? a1y : a2y;
        float gx = tg[2] * (float)W;
        float gy = tg[3] * (float)H;
        float gw = tg[4] * (float)W;
        float gh = tg[5] * (float)H;
        float rx = gw / (ax + 1e-16f);
        float ry = gh / (ay + 1e-16f);
        float mr = fmaxf(fmaxf(rx, 1.0f / rx), fmaxf(ry, 1.0f / ry));
        bool anchor_ok = mr < 4.0f;
        float gix_f = floorf(gx), giy_f = floorf(gy);
        float fx = gx - gix_f, fy = gy - giy_f;
        bool sel;
        float offx = 0.0f, offy = 0.0f;
        switch (off) {
            case 0: sel = true; break;
            case 1: sel = (fx > 0.5f) && (gix_f < (float)(W - 1)); offx = 1.0f; break;
            case 2: sel = (fy > 0.5f) && (giy_f < (float)(H - 1)); offy = 1.0f; break;
            case 3: sel = (fx < 0.5f) && (gix_f > 0.0f); offx = -1.0f; break;
            default: sel = (fy < 0.5f) && (giy_f > 0.0f); offy = -1.0f; break;
        }
        float vm = (sel && anchor_ok) ? 1.0f : 0.0f;
        float gex = gx - offx, gey = gy - offy;
        int gi = min(max((int)floorf(gex), 0), W - 1);
        int gj = min(max((int)floorf(gey), 0), H - 1);
        float tx = gex - (float)gi, ty = gey - (float)gj;
        float tx1 = tx - gw * 0.5f, ty1 = ty - gh * 0.5f;
        float tx2 = tx + gw * 0.5f, ty2 = ty + gh * 0.5f;
        int bi = (int)tg[0];
        int ci = (int)tg[1];
        const float* pm = pred + (size_t)((((bi * 3 + an) * H + gj) * W + gi)) * 85;
        float px = sigm_f(pm[0]) * 2.0f - 0.5f;
        float py = sigm_f(pm[1]) * 2.0f - 0.5f;
        float sw = sigm_f(pm[2]) * 2.0f;
        float sh = sigm_f(pm[3]) * 2.0f;
        float pw = sw * sw * ax;
        float ph = sh * sh * ay;
        float px1 = px - pw * 0.5f, py1 = py - ph * 0.5f;
        float px2 = px + pw * 0.5f, py2 = py + ph * 0.5f;
        float ix1 = fmaxf(px1, tx1), iy1 = fmaxf(py1, ty1);
        float ix2 = fminf(px2, tx2), iy2 = fminf(py2, ty2);
        float inter = fmaxf(ix2 - ix1, 0.0f) * fmaxf(iy2 - iy1, 0.0f);
        float area1 = (px2 - px1) * (py2 - py1);
        float area2 = (tx2 - tx1) * (ty2 - ty1);
        float uni = area1 + area2 - inter + 1e-7f;
        float iou = inter / uni;
        float ex1 = fminf(px1, tx1), ey1 = fminf(py1, ty1);
        float ex2 = fmaxf(px2, tx2), ey2 = fmaxf(py2, ty2);
        float c2 = (ex2 - ex1) * (ex2 - ex1) + (ey2 - ey1) * (ey2 - ey1) + 1e-7f;
        float dx = px1 + px2 - tx1 - tx2;
        float dy = py1 + py2 - ty1 - ty2;
        float rho2 = (dx * dx + dy * dy) * 0.25f;
        float w1 = px2 - px1, h1 = py2 - py1;
        float w2 = tx2 - tx1, h2 = ty2 - ty1;
        const float k4pi2 = (float)(4.0 / (3.14159265358979323846 * 3.14159265358979323846));
        float dat = atanf(w2 / (h2 + 1e-7f)) - atanf(w1 / (h1 + 1e-7f));
        float vv = k4pi2 * dat * dat;
        float alpha = vv / (1.0f - iou + vv + 1e-7f);
        float ciou = iou - rho2 / c2 - alpha * vv;
        lbox_c = (1.0f - ciou) * vm;
        cnt_c  = vm;
        float lc = 0.0f;
        for (int c = 0; c < 80; ++c)
            lc += bce_f(pm[5 + c], (c == ci) ? 1.0f : 0.0f);
        lcls_c = lc * vm;
        eCell[e] = ((bi * 3 + an) * H + gj) * W + gi;
        eVal[e]  = fmaxf(ciou, 0.0f) * vm;
        ePm4[e]  = pm[4];
    }
    float s0 = block_sum(lbox_c, sm);
    float s1 = block_sum(cnt_c, sm);
    float s2 = block_sum(lcls_c, sm);
    if (threadIdx.x == 0) {
        aPart[blockIdx.x * 3 + 0] = s0;
        aPart[blockIdx.x * 3 + 1] = s1;
        aPart[blockIdx.x * 3 + 2] = s2;
    }
}

// ---------------------------------------------------------------------------
// Kernel B: dedup + scatter-max in LDS; lobj correction = -x*max per cell.
// One block per layer (blockIdx.x = layer). Fully deterministic.
// ---------------------------------------------------------------------------
__global__ void objCorrKernel(const int* __restrict__ eCell,
                              const float* __restrict__ eVal,
                              const float* __restrict__ ePm4,
                              float* __restrict__ bRes, int nEnt)
{
    __shared__ int   sCell[ENT_MAX];
    __shared__ float sVal[ENT_MAX];
    __shared__ float sPm4[ENT_MAX];
    __shared__ float sm[8];
    int l = blockIdx.x;
    int n = min(nEnt, ENT_MAX);
    const int*   ec = eCell + l * nEnt;
    const float* ev = eVal  + l * nEnt;
    const float* ep = ePm4  + l * nEnt;
    for (int e = threadIdx.x; e < n; e += blockDim.x) {
        sCell[e] = ec[e];
        sVal[e]  = ev[e];
        sPm4[e]  = ep[e];
    }
    __syncthreads();
    float corr = 0.0f;
    for (int e = threadIdx.x; e < n; e += blockDim.x) {
        int cell = sCell[e];
        bool first = true;
        for (int j = 0; j < e; ++j)
            if (sCell[j] == cell) { first = false; break; }
        if (!first) continue;
        float m = sVal[e];
        for (int j = e + 1; j < n; ++j)
            if (sCell[j] == cell) m = fmaxf(m, sVal[j]);
        if (m > 0.0f) corr -= sPm4[e] * m; // bce(x,m) - bce(x,0) = -x*m
    }
    float s = block_sum(corr, sm);
    if (threadIdx.x == 0) bRes[l] = s;
}

// ---------------------------------------------------------------------------
// Kernel C: sum of bce(pred[...,4], 0). WMMA f32 16x16x4 accumulator
// (A = 64 bce values, B = ones, D += A*B). Fast path: full 256-cell chunks,
// no per-lane predicates (no exec manipulation), 8 NT loads off one base
// address with immediate offsets -> single clause, 256 lines in flight/wave.
// Tail: at most one partial chunk, wave-uniform branch.
// ---------------------------------------------------------------------------
__global__ void objBaseKernel(const float* __restrict__ pred,
                              float* __restrict__ cPart, int N)
{
    __shared__ float sm[8];
    const int lane = threadIdx.x & 31;
    const int wid  = threadIdx.x >> 5;
    const int gw   = __builtin_amdgcn_readfirstlane(
                        blockIdx.x * (blockDim.x >> 5) + wid);
    const int nw   = gridDim.x * (blockDim.x >> 5);
    const int CH   = 256;            // cells per wave-iteration (8 per lane)
    const int step = nw * CH;

    v8f acc = {0.0f, 0.0f, 0.0f, 0.0f, 0.0f, 0.0f, 0.0f, 0.0f};
    v2f ones; ones[0] = 1.0f; ones[1] = 1.0f;

    int i = gw * CH;
    for (; i + CH <= N; i += step) {
        const float* p = pred + (size_t)(i + lane) * 85 + 4;
        float x0 = __builtin_nontemporal_load(p + 0 * 32 * 85);
        float x1 = __builtin_nontemporal_load(p + 1 * 32 * 85);
        float x2 = __builtin_nontemporal_load(p + 2 * 32 * 85);
        float x3 = __builtin_nontemporal_load(p + 3 * 32 * 85);
        float x4 = __builtin_nontemporal_load(p + 4 * 32 * 85);
        float x5 = __builtin_nontemporal_load(p + 5 * 32 * 85);
        float x6 = __builtin_nontemporal_load(p + 6 * 32 * 85);
        float x7 = __builtin_nontemporal_load(p + 7 * 32 * 85);
        v2f a0; a0[0] = bce0_f(x0); a0[1] = bce0_f(x1);
        v2f a1; a1[0] = bce0_f(x2); a1[1] = bce0_f(x3);
        v2f a2; a2[0] = bce0_f(x4); a2[1] = bce0_f(x5);
        v2f a3; a3[0] = bce0_f(x6); a3[1] = bce0_f(x7);
        acc = __builtin_amdgcn_wmma_f32_16x16x4_f32(false, a0, false, ones, (short)0, acc, false, false);
        acc = __builtin_amdgcn_wmma_f32_16x16x4_f32(false, a1, false, ones, (short)0, acc, false, false);
        acc = __builtin_amdgcn_wmma_f32_16x16x4_f32(false, a2, false, ones, (short)0, acc, false, false);
        acc = __builtin_amdgcn_wmma_f32_16x16x4_f32(false, a3, false, ones, (short)0, acc, false, false);
    }
    if (i < N) { // wave-uniform tail: one partial chunk
        for (int k = 0; k < 8; k += 2) {
            int j0 = i + k * 32 + lane;
            int j1 = j0 + 32;
            float x0 = __builtin_nontemporal_load(pred + (size_t)min(j0, N - 1) * 85 + 4);
            float x1 = __builtin_nontemporal_load(pred + (size_t)min(j1, N - 1) * 85 + 4);
            v2f a;
            a[0] = (j0 < N) ? bce0_f(x0) : 0.0f;
            a[1] = (j1 < N) ? bce0_f(x1) : 0.0f;
            acc = __builtin_amdgcn_wmma_f32_16x16x4_f32(false, a, false, ones, (short)0, acc, false, false);
        }
    }
    float s = acc[0] + acc[1] + acc[2] + acc[3] + acc[4] + acc[5] + acc[6] + acc[7];
    // every column of D duplicates the row sums; take column 0 (lanes 0 & 16)
    float tot = __shfl(s, 0, 32) + __shfl(s, 16, 32);
    __syncthreads();
    if (lane == 0) sm[wid] = tot;
    __syncthreads();
    if (threadIdx.x == 0) {
        float r = 0.0f;
        int nwb = blockDim.x >> 5;
        for (int i2 = 0; i2 < nwb; ++i2) r += sm[i2];
        cPart[blockIdx.x] = r;
    }
}

// ---------------------------------------------------------------------------
// Final combine (single thread, fixed summation order).
// ---------------------------------------------------------------------------
__global__ void finalizeKernel(const float* __restrict__ aPart,
                               const float* __restrict__ bRes,
                               const float* __restrict__ cPart,
                               float* __restrict__ out,
                               int gridA, int ncell0, int ncell1, int ncell2,
                               int g0, int g1, int g2, int Bsz)
{
    if (threadIdx.x != 0 || blockIdx.x != 0) return;
    const float balance[3] = {4.0f, 1.0f, 0.4f};
    const int ncells[3] = {ncell0, ncell1, ncell2};
    const int gc[3] = {g0, g1, g2};
    float lb = 0.0f, lo = 0.0f, lc = 0.0f;
    for (int l = 0; l < 3; ++l) {
        float sl = 0.0f, sc = 0.0f, scl = 0.0f;
        for (int b = 0; b < gridA; ++b) {
            sl  += aPart[(l * gridA + b) * 3 + 0];
            sc  += aPart[(l * gridA + b) * 3 + 1];
            scl += aPart[(l * gridA + b) * 3 + 2];
        }
        float sbase = 0.0f;
        for (int b = 0; b < gc[l]; ++b) sbase += cPart[l * CPART_MAX + b];
        float lobj = (sbase + bRes[l]) / (float)ncells[l] * balance[l];
        float lbox = (sc > 0.0f) ? sl / fmaxf(sc, 1.0f) : 0.0f;
        float lcls = (sc > 0.0f) ? scl / fmaxf(sc * 80.0f, 1.0f) : 0.0f;
        lb += lbox; lo += lobj; lc += lcls;
    }
    out[0] = (0.05f * lb + 1.0f * lo + 0.5f * lc) * (float)Bsz;
}

static int gridC_for(int N) {
    int g = (N + 2047) / 2048; // 2048 cells per block per pass
    if (g < 1) g = 1;
    if (g > CPART_MAX) g = CPART_MAX;
    return g;
}

extern "C" void kernel_launch(void* const* d_in, const int* in_sizes, int n_in,
                              void* d_out, int out_size, void* d_ws, size_t ws_size,
                              hipStream_t stream)
{
    (void)n_in; (void)out_size; (void)ws_size;
    const float* pred0   = (const float*)d_in[0];
    const float* pred1   = (const float*)d_in[1];
    const float* pred2   = (const float*)d_in[2];
    const float* targets = (const float*)d_in[3];

    int NT   = in_sizes[3] / 6;                  // 300
    int B    = in_sizes[0] / (3 * 80 * 80 * 85); // 16
    int nEnt = 15 * NT;                          // 4500
    int gridA = (nEnt + 255) / 256;              // 18

    // workspace layout (4-byte slots): all regions written before read
    float* ws    = (float*)d_ws;
    int*   eCell = (int*)ws;                  // 3*nEnt ints
    float* eVal  = ws + 3 * nEnt;             // 3*nEnt floats
    float* ePm4  = ws + 6 * nEnt;             // 3*nEnt floats
    float* aPart = ws + 9 * nEnt;             // 3*gridA*3 floats
    float* bRes  = aPart + 3 * gridA * 3;     // 3 floats
    float* cPart = bRes + 3;                  // 3*CPART_MAX floats

    // ANCHORS / STRIDES (all exact binary fractions)
    const float A0[6] = {1.25f, 1.625f, 2.0f, 3.75f, 4.125f, 2.875f};
    const float A1[6] = {1.875f, 3.8125f, 3.875f, 2.8125f, 3.6875f, 7.4375f};
    const float A2[6] = {3.625f, 2.8125f, 4.875f, 6.1875f, 11.65625f, 10.1875f};

    assignKernel<<<gridA, 256, 0, stream>>>(pred0, targets,
        eCell + 0 * nEnt, eVal + 0 * nEnt, ePm4 + 0 * nEnt, aPart + 0 * gridA * 3,
        NT, 80, 80, A0[0], A0[1], A0[2], A0[3], A0[4], A0[5]);
    assignKernel<<<gridA, 256, 0, stream>>>(pred1, targets,
        eCell + 1 * nEnt, eVal + 1 * nEnt, ePm4 + 1 * nEnt, aPart + 1 * gridA * 3,
        NT, 40, 40, A1[0], A1[1], A1[2], A1[3], A1[4], A1[5]);
    assignKernel<<<gridA, 256, 0, stream>>>(pred2, targets,
        eCell + 2 * nEnt, eVal + 2 * nEnt, ePm4 + 2 * nEnt, aPart + 2 * gridA * 3,
        NT, 20, 20, A2[0], A2[1], A2[2], A2[3], A2[4], A2[5]);

    objCorrKernel<<<3, 256, 0, stream>>>(eCell, eVal, ePm4, bRes, nEnt);

    int n0 = B * 3 * 80 * 80;
    int n1 = B * 3 * 40 * 40;
    int n2 = B * 3 * 20 * 20;
    int g0 = gridC_for(n0), g1 = gridC_for(n1), g2 = gridC_for(n2);
    objBaseKernel<<<g0, 256, 0, stream>>>(pred0, cPart + 0 * CPART_MAX, n0);
    objBaseKernel<<<g1, 256, 0, stream>>>(pred1, cPart + 1 * CPART_MAX, n1);
    objBaseKernel<<<g2, 256, 0, stream>>>(pred2, cPart + 2 * CPART_MAX, n2);

    finalizeKernel<<<1, 1, 0, stream>>>(aPart, bRes, cPart, (float*)d_out,
                                        gridA, n0, n1, n2, g0, g1, g2, B);
}